// GNNModel_88845693485829
// MI455X (gfx1250) — compile-verified
//
#include <hip/hip_runtime.h>

#define NN 15000
#define EE 60000
#define KBIG 4096      // NODE_DIM*HID
#define HEADS_W 240    // 16+32+64+128

typedef __attribute__((ext_vector_type(2))) float v2f;
typedef __attribute__((ext_vector_type(8))) float v8f;

// ---------------- small prep kernels ----------------

__global__ void k_zero(float* __restrict__ p, int n) {
  int i = blockIdx.x * blockDim.x + threadIdx.x;
  if (i < n) p[i] = 0.f;
}

// w2T[i*64+k][o] = w2[k][i*64+o]   (w2 is [64,4096] row-major)
__global__ void k_w2t(const float* __restrict__ w2, float* __restrict__ w2T) {
  int idx = blockIdx.x * blockDim.x + threadIdx.x;  // over 4096*64
  if (idx >= KBIG * 64) return;
  int j = idx >> 6;           // row of w2T = i*64+k
  int o = idx & 63;
  int i = j >> 6, k = j & 63;
  w2T[idx] = w2[k * KBIG + i * 64 + o];
}

// pack 4 head weight matrices into wH[64][240] and biases into hb[240]
__global__ void k_packheads(const float* __restrict__ ws_, const float* __restrict__ bs_,
                            const float* __restrict__ wn_, const float* __restrict__ bn_,
                            const float* __restrict__ wt_, const float* __restrict__ bt_,
                            const float* __restrict__ wp_, const float* __restrict__ bp_,
                            float* __restrict__ wH, float* __restrict__ hb) {
  int idx = blockIdx.x * blockDim.x + threadIdx.x;
  if (idx < 64 * HEADS_W) {
    int k = idx / HEADS_W, c = idx % HEADS_W;
    float v;
    if (c < 16)       v = ws_[k * 16 + c];
    else if (c < 48)  v = wn_[k * 32 + (c - 16)];
    else if (c < 112) v = wt_[k * 64 + (c - 48)];
    else              v = wp_[k * 128 + (c - 112)];
    wH[idx] = v;
  } else if (idx < 64 * HEADS_W + HEADS_W) {
    int c = idx - 64 * HEADS_W;
    float v;
    if (c < 16)       v = bs_[c];
    else if (c < 48)  v = bn_[c - 16];
    else if (c < 112) v = bt_[c - 48];
    else              v = bp_[c - 112];
    hb[c] = v;
  }
}

// ---------------- node features + per-node GEMMs ----------------
// xf[n,64] = [x | inEmb | outEmb], rootv = xf@root_w + conv_b, biasv = xf@reshape(b2,[64,64])
__global__ void __launch_bounds__(256) k_node(
    const float* __restrict__ x,
    const int* __restrict__ inp, const int* __restrict__ outp,
    const float* __restrict__ inEmb, const float* __restrict__ outEmb,
    const float* __restrict__ rootW, const float* __restrict__ convB,
    const float* __restrict__ b2,
    float* __restrict__ xf, float* __restrict__ rootv, float* __restrict__ biasv) {
  __shared__ float rwL[64 * 64];
  __shared__ float b2L[64 * 64];
  __shared__ float cbL[64];
  __shared__ float xfL[32][64];
  int tid = threadIdx.x;
  int nb = blockIdx.x * 32;
  for (int idx = tid; idx < 4096; idx += 256) { rwL[idx] = rootW[idx]; b2L[idx] = b2[idx]; }
  if (tid < 64) cbL[tid] = convB[tid];
  for (int idx = tid; idx < 32 * 64; idx += 256) {
    int nl = idx >> 6, f = idx & 63;
    int n = nb + nl;
    float v = 0.f;
    if (n < NN) {
      if (f < 32)      v = x[n * 32 + f];
      else if (f < 48) v = inEmb[inp[n] * 16 + (f - 32)];
      else             v = outEmb[outp[n] * 16 + (f - 48)];
      xf[n * 64 + f] = v;
    }
    xfL[nl][f] = v;
  }
  __syncthreads();
  for (int idx = tid; idx < 32 * 64; idx += 256) {
    int nl = idx >> 6, o = idx & 63;
    int n = nb + nl;
    if (n >= NN) continue;
    float ar = cbL[o], ab = 0.f;
#pragma unroll 8
    for (int i = 0; i < 64; ++i) {
      float xv = xfL[nl][i];
      ar += xv * rwL[i * 64 + o];
      ab += xv * b2L[i * 64 + o];
    }
    rootv[n * 64 + o] = ar;
    biasv[n * 64 + o] = ab;
  }
}

// ---------------- edge MLP layer 1 + degree counts ----------------
__global__ void __launch_bounds__(256) k_h(
    const int* __restrict__ ei,   // [2,E]
    const int* __restrict__ ent, const int* __restrict__ enp,
    const float* __restrict__ esc,
    const float* __restrict__ ntEmb, const float* __restrict__ npEmb,
    const float* __restrict__ w1, const float* __restrict__ b1,
    float* __restrict__ hbuf, float* __restrict__ cnt) {
  __shared__ float w1L[33 * 64];
  __shared__ float b1L[64];
  __shared__ float eaL[64][34];
  int tid = threadIdx.x;
  int eb = blockIdx.x * 64;
  for (int idx = tid; idx < 33 * 64; idx += 256) w1L[idx] = w1[idx];
  if (tid < 64) b1L[tid] = b1[tid];
  for (int idx = tid; idx < 64 * 33; idx += 256) {
    int el = idx / 33, j = idx % 33;
    int e = eb + el;
    float v = 0.f;
    if (e < EE) {
      if (j < 16)      v = ntEmb[ent[e] * 16 + j];
      else if (j < 32) v = npEmb[enp[e] * 16 + (j - 16)];
      else             v = esc[e];
    }
    eaL[el][j] = v;
  }
  if (tid < 64) {
    int e = eb + tid;
    if (e < EE) atomicAdd(&cnt[ei[EE + e]], 1.0f);
  }
  __syncthreads();
  for (int idx = tid; idx < 64 * 64; idx += 256) {
    int el = idx >> 6, o = idx & 63;
    int e = eb + el;
    if (e >= EE) continue;
    float a = b1L[o];
#pragma unroll
    for (int j = 0; j < 33; ++j) a += eaL[el][j] * w1L[j * 64 + o];
    hbuf[e * 64 + o] = fmaxf(a, 0.f);
  }
}

// ---------------- fused bilinear message GEMM (WMMA f32) + scatter ----------------
// msg[e,o] = sum_{i,k} xf[src_e,i]*h[e,k]*w2T[i*64+k][o]  ; 8 waves/block, 16 edges/wave
__global__ void __launch_bounds__(256) k_msg(
    const int* __restrict__ ei,
    const float* __restrict__ xf,
    const float* __restrict__ hbuf,
    const float* __restrict__ w2T,
    const float* __restrict__ biasv,
    float* __restrict__ sums) {
  __shared__ __align__(16) float hL[8][16][64];   // 32 KB: per-wave h tile
  __shared__ __align__(16) float Bs[32][64][2];   // 16 KB: (row2k, row2k+1) pairs for current i
  __shared__ int srcL[8][16];
  int tid = threadIdx.x;
  int w = tid >> 5;
  int l = tid & 31;
  int te = (blockIdx.x * 8 + w) * 16;

  for (int idx = l; idx < 16 * 64; idx += 32) {
    int m = idx >> 6, f = idx & 63;
    int e = te + m;
    hL[w][m][f] = (e < EE) ? hbuf[e * 64 + f] : 0.f;   // zero rows kill tail edges
  }
  if (l < 16) {
    int e = te + l;
    srcL[w][l] = (e < EE) ? ei[e] : 0;
  }
  __syncthreads();

  int m0 = l & 15;                 // A row / B col-in-tile for this lane
  int halfk = (l >> 4) << 1;       // K sub-offset 0 or 2
  long sA = (long)srcL[w][m0] * 64;

  v8f acc[4] = {};
  for (int i = 0; i < 64; ++i) {
    __syncthreads();
    for (int idx = tid; idx < 4096; idx += 256) {
      int k = idx >> 6, o = idx & 63;
      Bs[k >> 1][o][k & 1] = w2T[(i * 64 + k) * 64 + o];
      if (i < 63) __builtin_prefetch(&w2T[((i + 1) * 64 + k) * 64 + o], 0, 0);
    }
    __syncthreads();
    float xi = xf[sA + i];
#pragma unroll
    for (int ks = 0; ks < 16; ++ks) {
      int kbase = ks * 4 + halfk;
      v2f hp = *(const v2f*)&hL[w][m0][kbase];
      v2f a; a.x = xi * hp.x; a.y = xi * hp.y;   // Z row synthesized in registers
      int k2 = kbase >> 1;
      v2f bf[4];
      bf[0] = *(const v2f*)&Bs[k2][ 0 + m0][0];
      bf[1] = *(const v2f*)&Bs[k2][16 + m0][0];
      bf[2] = *(const v2f*)&Bs[k2][32 + m0][0];
      bf[3] = *(const v2f*)&Bs[k2][48 + m0][0];
#pragma unroll
      for (int t = 0; t < 4; ++t)
        acc[t] = __builtin_amdgcn_wmma_f32_16x16x4_f32(
            false, a, false, bf[t], (short)0, acc[t], false, false);
    }
  }

  // bias (xf_src @ reshape(b2)) + scatter-add into sums[dst]
  int rowOff = (l >> 4) << 3;      // 0 or 8
#pragma unroll
  for (int t = 0; t < 4; ++t) {
#pragma unroll
    for (int r = 0; r < 8; ++r) {
      int m = r + rowOff;
      int e = te + m;
      if (e < EE) {
        int o = t * 16 + m0;
        int s = srcL[w][m];
        float v = acc[t][r] + biasv[(long)s * 64 + o];
        int d = ei[EE + e];
        atomicAdd(&sums[(long)d * 64 + o], v);
      }
    }
  }
}

// ---------------- finalize + 4 heads as one packed WMMA GEMM ----------------
__global__ void __launch_bounds__(256) k_out(
    const float* __restrict__ sums, const float* __restrict__ cnt,
    const float* __restrict__ rootv,
    const float* __restrict__ wH, const float* __restrict__ hb,
    float* __restrict__ dout) {
  __shared__ __align__(16) float ov[8][16][64];  // relu'd node vectors
  int tid = threadIdx.x;
  int w = tid >> 5, l = tid & 31;
  int nb = blockIdx.x * 128 + w * 16;

  for (int idx = l; idx < 16 * 64; idx += 32) {
    int m = idx >> 6, o = idx & 63;
    int n = nb + m;
    float v = 0.f;
    if (n < NN) {
      float c = cnt[n];
      v = sums[n * 64 + o] / fmaxf(c, 1.f) + rootv[n * 64 + o];
      v = fmaxf(v, 0.f);
    }
    ov[w][m][o] = v;
  }
  __syncthreads();

  int m0 = l & 15;
  int halfk = (l >> 4) << 1;
  int rowOff = (l >> 4) << 3;
  for (int t = 0; t < 15; ++t) {          // 15 col-tiles over packed 240 outputs
    v8f acc = {};
    int c = t * 16 + m0;
#pragma unroll
    for (int ks = 0; ks < 16; ++ks) {
      int kbase = ks * 4 + halfk;
      v2f a = *(const v2f*)&ov[w][m0][kbase];
      v2f b; b.x = wH[kbase * HEADS_W + c]; b.y = wH[(kbase + 1) * HEADS_W + c];
      acc = __builtin_amdgcn_wmma_f32_16x16x4_f32(
          false, a, false, b, (short)0, acc, false, false);
    }
    int segStart, width; long base;
    if (t == 0)      { segStart = 0;   width = 16;  base = 0; }
    else if (t < 3)  { segStart = 16;  width = 32;  base = (long)NN * 16; }
    else if (t < 7)  { segStart = 48;  width = 64;  base = (long)NN * 48; }
    else             { segStart = 112; width = 128; base = (long)NN * 112; }
    float bias = hb[c];
#pragma unroll
    for (int r = 0; r < 8; ++r) {
      int n = nb + r + rowOff;
      if (n < NN) dout[base + (long)n * width + (c - segStart)] = acc[r] + bias;
    }
  }
}

// ---------------- launcher ----------------
extern "C" void kernel_launch(void* const* d_in, const int* in_sizes, int n_in,
                              void* d_out, int out_size, void* d_ws, size_t ws_size,
                              hipStream_t stream) {
  (void)in_sizes; (void)n_in; (void)out_size; (void)ws_size;
  const float* x           = (const float*)d_in[0];
  const int*   input_np    = (const int*)d_in[1];
  const int*   output_np   = (const int*)d_in[2];
  const int*   edge_index  = (const int*)d_in[3];
  const int*   edge_nt     = (const int*)d_in[4];
  const int*   edge_np     = (const int*)d_in[5];
  const float* edge_sc     = (const float*)d_in[6];
  const float* in_np_emb   = (const float*)d_in[7];
  const float* out_np_emb  = (const float*)d_in[8];
  const float* edge_np_emb = (const float*)d_in[9];
  const float* edge_nt_emb = (const float*)d_in[10];
  const float* mlp_w1      = (const float*)d_in[11];
  const float* mlp_b1      = (const float*)d_in[12];
  const float* mlp_w2      = (const float*)d_in[13];
  const float* mlp_b2      = (const float*)d_in[14];
  const float* root_w      = (const float*)d_in[15];
  const float* conv_b      = (const float*)d_in[16];
  const float* w_super     = (const float*)d_in[17];
  const float* b_super     = (const float*)d_in[18];
  const float* w_nt        = (const float*)d_in[19];
  const float* b_nt        = (const float*)d_in[20];
  const float* w_tags      = (const float*)d_in[21];
  const float* b_tags      = (const float*)d_in[22];
  const float* w_primary   = (const float*)d_in[23];
  const float* b_primary   = (const float*)d_in[24];
  float* out = (float*)d_out;

  float* ws = (float*)d_ws;
  size_t off = 0;
  float* xf    = ws + off; off += (size_t)NN * 64;
  float* rootv = ws + off; off += (size_t)NN * 64;
  float* biasv = ws + off; off += (size_t)NN * 64;
  float* sums  = ws + off; off += (size_t)NN * 64;
  float* cnt   = ws + off; off += (size_t)NN;        // contiguous after sums
  float* hbuf  = ws + off; off += (size_t)EE * 64;
  float* w2T   = ws + off; off += (size_t)KBIG * 64;
  float* wH    = ws + off; off += (size_t)64 * HEADS_W;
  float* hb    = ws + off; off += (size_t)HEADS_W;

  k_zero<<<(NN * 65 + 255) / 256, 256, 0, stream>>>(sums, NN * 65);
  k_w2t<<<(KBIG * 64 + 255) / 256, 256, 0, stream>>>(mlp_w2, w2T);
  k_packheads<<<(64 * HEADS_W + HEADS_W + 255) / 256, 256, 0, stream>>>(
      w_super, b_super, w_nt, b_nt, w_tags, b_tags, w_primary, b_primary, wH, hb);
  k_node<<<(NN + 31) / 32, 256, 0, stream>>>(
      x, input_np, output_np, in_np_emb, out_np_emb, root_w, conv_b, mlp_b2,
      xf, rootv, biasv);
  k_h<<<(EE + 63) / 64, 256, 0, stream>>>(
      edge_index, edge_nt, edge_np, edge_sc, edge_nt_emb, edge_np_emb,
      mlp_w1, mlp_b1, hbuf, cnt);
  k_msg<<<(EE + 127) / 128, 256, 0, stream>>>(edge_index, xf, hbuf, w2T, biasv, sums);
  k_out<<<(NN + 127) / 128, 256, 0, stream>>>(sums, cnt, rootv, wH, hb, out);
}